// PolicyClassifier_65429531787382
// MI455X (gfx1250) — compile-verified
//
#include <hip/hip_runtime.h>
#include <hip/hip_bf16.h>

#define W_FEAT 5
#define WIN 11
#define RPB 256
#define LDS_N (RPB + 2 * W_FEAT)

typedef __attribute__((ext_vector_type(2))) float v2f;
typedef __attribute__((ext_vector_type(8))) float v8f;

// lane <-> lane^16 exchange via DS_SWIZZLE (xor_mask=0x10, and_mask=0x1f)
__device__ __forceinline__ float swap16(float x) {
  return __int_as_float(__builtin_amdgcn_ds_swizzle(__float_as_int(x), 0x401f));
}

#define CE(i, j)                   \
  do {                             \
    float _lo = fminf(s[i], s[j]); \
    float _hi = fmaxf(s[i], s[j]); \
    s[i] = _lo;                    \
    s[j] = _hi;                    \
  } while (0)

// Full optimal 35-CE sorting network for n=11 (edge path: needs s[2..10]).
__device__ __forceinline__ void sort11_full(float (&s)[WIN]) {
  CE(0, 9); CE(1, 6); CE(2, 4); CE(3, 7); CE(5, 8);
  CE(0, 1); CE(3, 5); CE(4, 10); CE(6, 9); CE(7, 8);
  CE(1, 3); CE(2, 5); CE(4, 7); CE(8, 10);
  CE(0, 4); CE(1, 2); CE(3, 7); CE(5, 9); CE(6, 8);
  CE(0, 1); CE(2, 6); CE(4, 5); CE(7, 8); CE(9, 10);
  CE(2, 4); CE(3, 6); CE(5, 7); CE(8, 9);
  CE(1, 2); CE(3, 4); CE(5, 6); CE(7, 8);
  CE(2, 3); CE(4, 5); CE(6, 7);
}

// Cone-of-influence pruned network (29 CE): positions 5 (median) and 10 (max)
// are bit-identical to the full sort; other positions are unused here.
__device__ __forceinline__ void sort11_med_max(float (&s)[WIN]) {
  CE(0, 9); CE(1, 6); CE(2, 4); CE(3, 7); CE(5, 8);
  CE(0, 1); CE(3, 5); CE(4, 10); CE(6, 9); CE(7, 8);
  CE(1, 3); CE(2, 5); CE(4, 7); CE(8, 10);
  CE(0, 4); CE(1, 2); CE(3, 7); CE(5, 9); CE(6, 8);
  CE(2, 6); CE(4, 5); CE(7, 8); CE(9, 10);
  CE(2, 4); CE(3, 6); CE(5, 7);
  CE(3, 4); CE(5, 6);
  CE(4, 5);
}
#undef CE

template <bool EDGE>
__device__ __forceinline__ void process_rows(
    const float* __restrict__ stage, int tid, int cbase, int L,
    const float* __restrict__ w1, const float* __restrict__ b1,
    const float* __restrict__ w2, float bias2, float* __restrict__ out) {
  const int lane = tid & 31;
  const int wv = tid >> 5;
  const int nl = lane & 15;
  const int half = lane >> 4;  // 0: lanes 0-15, 1: lanes 16-31
  const int c = cbase + tid;   // this lane's window center

  // ---- window from LDS ----
  float g[WIN];
#pragma unroll
  for (int j = 0; j < WIN; ++j) g[j] = stage[tid + j];

  int nv = WIN;
  if (EDGE) {
    const int lo = c - W_FEAT, hi = c + W_FEAT;
    nv = (hi >= L ? L - 1 : hi) - (lo < 0 ? 0 : lo) + 1;
  }

  // ---- sort (edge: invalid -> +inf, matches reference) ----
  float s[WIN];
  const float INF = __builtin_inff();
#pragma unroll
  for (int j = 0; j < WIN; ++j) {
    if (EDGE) {
      int idx = c - W_FEAT + j;
      s[j] = (idx >= 0 && idx < L) ? g[j] : INF;
    } else {
      s[j] = g[j];
    }
  }

  float med, peak;
  if (EDGE) {
    sort11_full(s);
    const int k = (nv - 1) >> 1;  // lower median index
    med = (k == 2) ? s[2] : (k == 3) ? s[3] : (k == 4) ? s[4] : s[5];
    peak = (nv <= 6) ? s[5]
         : (nv == 7) ? s[6]
         : (nv == 8) ? s[7]
         : (nv == 9) ? s[8]
         : (nv == 10) ? s[9] : s[10];
  } else {
    sort11_med_max(s);
    med = s[5];
    peak = s[10];
  }

  int dcnt = 0;
  float sum = 0.0f;
#pragma unroll
  for (int j = 0; j < WIN; ++j) {
    bool valid = true;
    if (EDGE) {
      int idx = c - W_FEAT + j;
      valid = (idx >= 0 && idx < L);
    }
    dcnt += (valid && (g[j] > med)) ? 1 : 0;
    sum += valid ? g[j] : 0.0f;
  }
  const float mean = sum / (float)nv;
  const float gam = peak / fmaxf(mean, 1e-8f);
  const float fdur = (float)dcnt;

  // ---- B = feats^T (4x16) per 16-row tile; B[k,m]: vgpr v, lane -> k=v+2*half
  // feats K-order: [dur, gamma, peak, mean]; stats of row (wave_base+r) live in lane r.
  const float pk_lo = __shfl(peak, nl, 32);        // peak of rows 0..15
  const float mn_lo = __shfl(mean, nl, 32);        // mean of rows 0..15
  const float du_hi = __shfl(fdur, 16 | nl, 32);   // dur  of rows 16..31
  const float ga_hi = __shfl(gam, 16 | nl, 32);    // gamma of rows 16..31
  v2f B0, B1;
  B0.x = half ? pk_lo : fdur;  // tile0: K=0 (dur) / K=2 (peak)
  B0.y = half ? mn_lo : gam;   // tile0: K=1 (gamma) / K=3 (mean)
  B1.x = half ? peak : du_hi;  // tile1
  B1.y = half ? mean : ga_hi;

  // ---- A = W1 (16 hidden x 4 feats) per hidden-half; C = b1 slice ----
  const int h8 = half << 3;    // 0 or 8
  const int k2 = half << 1;    // 0 or 2
  const v2f A0 = *(const v2f*)(w1 + nl * 4 + k2);         // hidden 0..15
  const v2f A1 = *(const v2f*)(w1 + (16 + nl) * 4 + k2);  // hidden 16..31
  const v8f C0 = *(const v8f*)(b1 + h8);
  const v8f C1 = *(const v8f*)(b1 + 16 + h8);
  const v8f w2a = *(const v8f*)(w2 + h8);
  const v8f w2b = *(const v8f*)(w2 + 16 + h8);

  // ---- layer 1: h^T = W1 @ feats^T + b1 via V_WMMA_F32_16X16X4_F32 ----
  // D_tH[n', m] = h[tile_t row m, H + n'] ; lane holds hidden n'=rr+8*half along vgprs
  v8f D00 = __builtin_amdgcn_wmma_f32_16x16x4_f32(false, A0, false, B0,
                                                  (short)0, C0, false, false);
  v8f D01 = __builtin_amdgcn_wmma_f32_16x16x4_f32(false, A1, false, B0,
                                                  (short)0, C1, false, false);
  v8f D10 = __builtin_amdgcn_wmma_f32_16x16x4_f32(false, A0, false, B1,
                                                  (short)0, C0, false, false);
  v8f D11 = __builtin_amdgcn_wmma_f32_16x16x4_f32(false, A1, false, B1,
                                                  (short)0, C1, false, false);

  // ---- layer 2: in-register relu-dot over 16 hidden units per lane ----
  float acc0 = 0.0f, acc1 = 0.0f;
#pragma unroll
  for (int rr = 0; rr < 8; ++rr) {
    acc0 += fmaxf(D00[rr], 0.0f) * w2a[rr] + fmaxf(D01[rr], 0.0f) * w2b[rr];
    acc1 += fmaxf(D10[rr], 0.0f) * w2a[rr] + fmaxf(D11[rr], 0.0f) * w2b[rr];
  }
  // combine hidden halves across lane pairs (lane <-> lane^16)
  const float full0 = acc0 + swap16(acc0);  // logit of row wave_base + nl
  const float full1 = acc1 + swap16(acc1);  // logit of row wave_base + 16 + nl
  // lane l writes row wave_base + l: fully coalesced 32-lane store
  out[1 + cbase + (wv << 5) + lane] = (half ? full1 : full0) + bias2;
}

__global__ __launch_bounds__(RPB) void policy_classifier_kernel(
    const float* __restrict__ delta_s, const float* __restrict__ w1,
    const float* __restrict__ b1, const float* __restrict__ w2,
    const float* __restrict__ b2, float* __restrict__ out, int L) {
  __shared__ float stage[LDS_N];
  const int tid = threadIdx.x;
  const int cbase = (int)blockIdx.x * RPB;

  for (int i = tid; i < LDS_N; i += RPB) {
    int gg = cbase - W_FEAT + i;
    gg = gg < 0 ? 0 : (gg >= L ? L - 1 : gg);
    stage[i] = delta_s[gg];
  }
  __syncthreads();

  const float bias2 = b2[0];
  // block-uniform scalar branch: EXEC stays all-ones inside (WMMA requirement)
  if (cbase >= W_FEAT && cbase + RPB + W_FEAT <= L) {
    process_rows<false>(stage, tid, cbase, L, w1, b1, w2, bias2, out);
  } else {
    process_rows<true>(stage, tid, cbase, L, w1, b1, w2, bias2, out);
  }

  if (blockIdx.x == 0 && tid == 0) out[0] = 0.0f;  // boundary_logits[0]
}

extern "C" void kernel_launch(void* const* d_in, const int* in_sizes, int n_in,
                              void* d_out, int out_size, void* d_ws,
                              size_t ws_size, hipStream_t stream) {
  const float* delta_s = (const float*)d_in[0];
  const float* w1 = (const float*)d_in[1];
  const float* b1 = (const float*)d_in[2];
  const float* w2 = (const float*)d_in[3];
  const float* b2 = (const float*)d_in[4];
  float* out = (float*)d_out;
  const int L = in_sizes[0];  // 4,000,000 (multiple of 256)
  const int blocks = (L + RPB - 1) / RPB;
  policy_classifier_kernel<<<blocks, RPB, 0, stream>>>(delta_s, w1, b1, w2, b2,
                                                       out, L);
}